// HybridSummaryEncoder_22428319220695
// MI455X (gfx1250) — compile-verified
//
#include <hip/hip_runtime.h>
#include <hip/hip_bf16.h>

typedef __attribute__((ext_vector_type(2))) float v2f;
typedef __attribute__((ext_vector_type(4))) float v4f;
typedef __attribute__((ext_vector_type(8))) float v8f;

#define B_     32
#define T_     4096
#define C_     512
#define NSEQ   4096   // cfg.max_seq_len (policy threshold)
#define NSPLIT 8      // T-dimension splits for the summary reduction

// Workspace layout (floats):
//   [0, B_*NSPLIT*C_)      : partial colsums  partial[(b*NSPLIT+s)*C_ + c]   (512 KB)
//   [B_*NSPLIT*C_, +B_*8)  : per-b scalars, stride 8:
//       +0 S_td (f32)  +1 Ndyn (f32)  +2 S_nv (f32)  +3 Nnvm (f32)
//       +4 first_dyn (i32 alias)      +5 use (i32 alias)

// ---------------------------------------------------------------------------
// Kernel 1: per-batch-row scalar reductions
// ---------------------------------------------------------------------------
__global__ __launch_bounds__(256) void hse_scalars_kernel(
    const unsigned char* __restrict__ static_mask,
    const unsigned char* __restrict__ nvm,
    const float* __restrict__ td,
    const float* __restrict__ nv,
    const unsigned char* __restrict__ mask,
    float* __restrict__ scal_f, int* __restrict__ scal_i) {
  const int b = blockIdx.x;
  const int tid = threadIdx.x;

  float s_td = 0.f, s_nv = 0.f;
  int ndyn = 0, nnvm = 0, lsum = 0, fmin = 0x7fffffff;

  for (int t = tid; t < T_; t += 256) {
    const int idx = b * T_ + t;
    const int dyn = static_mask[idx] ? 0 : 1;
    const int m   = nvm[idx] ? 1 : 0;
    ndyn += dyn;
    s_td += dyn ? td[idx] : 0.f;
    s_nv += (dyn & m) ? nv[idx] : 0.f;
    nnvm += dyn & m;
    lsum += mask[idx] ? 1 : 0;
    if (dyn && t < fmin) fmin = t;
  }

  __shared__ float shA[256], shB[256];
  __shared__ int   shC[256], shD[256], shE[256], shF[256];
  shA[tid] = s_td; shB[tid] = s_nv;
  shC[tid] = ndyn; shD[tid] = nnvm; shE[tid] = lsum; shF[tid] = fmin;
  __syncthreads();
  for (int s = 128; s > 0; s >>= 1) {
    if (tid < s) {
      shA[tid] += shA[tid + s];
      shB[tid] += shB[tid + s];
      shC[tid] += shC[tid + s];
      shD[tid] += shD[tid + s];
      shE[tid] += shE[tid + s];
      shF[tid] = min(shF[tid], shF[tid + s]);
    }
    __syncthreads();
  }
  if (tid == 0) {
    const int base = b * 8;
    scal_f[base + 0] = shA[0];                 // S_td
    scal_f[base + 1] = (float)shC[0];          // Ndyn
    scal_f[base + 2] = shB[0];                 // S_nv
    scal_f[base + 3] = (float)shD[0];          // Nnvm
    scal_i[base + 4] = (shF[0] == 0x7fffffff) ? 0 : shF[0];  // first_dyn (argmax semantics)
    scal_i[base + 5] = (shE[0] >= NSEQ) ? 1 : 0;             // use
  }
}

// ---------------------------------------------------------------------------
// Kernel 2: partial colsum over a T-slice (deterministic: disjoint slots,
// no atomics). Block (b, split) handles 512 tokens, LDS-staged codes.
// ---------------------------------------------------------------------------
__global__ __launch_bounds__(512) void hse_colsum_kernel(
    const int* __restrict__ code,
    const unsigned char* __restrict__ static_mask,
    const float* __restrict__ code_table,
    float* __restrict__ partial) {
  const int b     = blockIdx.x / NSPLIT;
  const int split = blockIdx.x % NSPLIT;
  const int c     = threadIdx.x;            // 512 threads == C_
  const int t0    = split * (T_ / NSPLIT);  // 512 tokens per block

  __shared__ int codeS[T_ / NSPLIT];
  __shared__ unsigned char dynS[T_ / NSPLIT];

  const int t = t0 + threadIdx.x;
  codeS[threadIdx.x] = code[b * T_ + t];
  dynS[threadIdx.x]  = static_mask[b * T_ + t] ? 0 : 1;
  __syncthreads();

  float acc = 0.f;
  #pragma unroll 4
  for (int j = 0; j < T_ / NSPLIT; ++j) {
    if (dynS[j]) acc += code_table[(size_t)codeS[j] * C_ + c];  // 2KB coalesced row read
  }
  partial[((size_t)b * NSPLIT + split) * C_ + c] = acc;
}

// ---------------------------------------------------------------------------
// Kernel 3: main embedding pass. One wave builds 16(c)x16(t) tiles with a
// single v_wmma_f32_16x16x4_f32 carrying ALL four rank-1 terms:
//   K0: dw (x) td*dyn     K1: vw (x) nv*nvm
//   K2: db (x) dyn        K3: vb (x) nvm
//   C : pure code_table gather tile
// A layout (16x4, M=lane%16): lanes 0-15 -> vgpr0=K0,vgpr1=K1;
//                             lanes 16-31 -> vgpr0=K2,vgpr1=K3.
// => per-lane base-pointer select makes A loads unconditional (no exec branches).
// B layout (4x16, N=lane%16): vgpr0: K0 (lo lanes) / K2 (hi lanes);
//                             vgpr1: K1 (lo lanes) / K3 (hi lanes).
// Output stores are non-temporal (write-once stream; keep L2 for code_table).
// ---------------------------------------------------------------------------
__global__ __launch_bounds__(256) void hse_embed_kernel(
    const int* __restrict__ code,
    const float* __restrict__ nv,
    const float* __restrict__ td,
    const unsigned char* __restrict__ static_mask,
    const unsigned char* __restrict__ nvm,
    const float* __restrict__ dw, const float* __restrict__ db,
    const float* __restrict__ vw, const float* __restrict__ vb,
    const float* __restrict__ code_table,
    float* __restrict__ out) {
  const int b    = blockIdx.x >> 5;        // 32 blocks per batch row
  const int tgrp = blockIdx.x & 31;        // each block covers 8 t-tiles
  const int wave = threadIdx.x >> 5;
  const int lane = threadIdx.x & 31;
  const int n    = lane & 15;              // tile-local column (t) / row (A: M)
  const int half = lane >> 4;

  const int t0 = (tgrp * 8 + wave) * 16;
  const int t  = t0 + n;
  const size_t bt = (size_t)b * T_ + t;

  const float tdv  = td[bt];
  const float nvv  = nv[bt];
  const float dynf = static_mask[bt] ? 0.f : 1.f;
  const float nvmf = nvm[bt] ? 1.f : 0.f;
  const int   cidx = code[bt];
  const float* crow = code_table + (size_t)cidx * C_;

  // B rows (per-lane, computed once): K0=td*dyn K1=nv*nvm K2=dyn K3=nvm
  v2f bvec;
  bvec.x = half ? dynf : (tdv * dynf);
  bvec.y = half ? nvmf : (nvv * nvmf);

  // A column sources (per-lane pointer select, once): lo lanes dw/vw, hi lanes db/vb
  const float* const pA0 = half ? db : dw;
  const float* const pA1 = half ? vb : vw;

  float* const obase = out + (size_t)b * C_ * T_ + t;

  for (int c0 = 0; c0 < C_; c0 += 16) {
    v2f avec;
    avec.x = pA0[c0 + n];   // unconditional b32 load
    avec.y = pA1[c0 + n];   // unconditional b32 load

    // This lane's 8 C/D rows are c = cl + r (contiguous in code_table row).
    const int cl = c0 + 8 * half;
    const v4f g0 = *(const v4f*)(crow + cl);
    const v4f g1 = *(const v4f*)(crow + cl + 4);

    v8f cacc;
    cacc[0] = g0.x; cacc[1] = g0.y; cacc[2] = g0.z; cacc[3] = g0.w;
    cacc[4] = g1.x; cacc[5] = g1.y; cacc[6] = g1.z; cacc[7] = g1.w;

    // D = A x B + C  (EXEC all-1s: fully uniform control flow in this kernel)
    v8f d = __builtin_amdgcn_wmma_f32_16x16x4_f32(
        /*neg_a=*/false, avec, /*neg_b=*/false, bvec,
        /*c_mod=*/(short)0, cacc, /*reuse_a=*/false, /*reuse_b=*/false);

    float* const op = obase + (size_t)cl * T_;
    #pragma unroll
    for (int r = 0; r < 8; ++r)
      __builtin_nontemporal_store(d[r], op + (size_t)r * T_);  // coalesced over 16 lanes
  }
}

// ---------------------------------------------------------------------------
// Kernel 4: finalize summary from partials + inject into column first_dyn.
// Touches only B_*C_ elements; runs after the embed pass on the same stream.
// ---------------------------------------------------------------------------
__global__ __launch_bounds__(512) void hse_inject_kernel(
    const float* __restrict__ partial,
    const float* __restrict__ dw, const float* __restrict__ db,
    const float* __restrict__ vw, const float* __restrict__ vb,
    const float* __restrict__ scal_f, const int* __restrict__ scal_i,
    float* __restrict__ out) {
  const int b = blockIdx.x;
  const int c = threadIdx.x;  // 512 threads == C_

  const int base = b * 8;
  if (!scal_i[base + 5]) return;  // use == 0: no injection for this row (uniform branch)

  float acc = 0.f;
  #pragma unroll
  for (int s = 0; s < NSPLIT; ++s)
    acc += partial[((size_t)b * NSPLIT + s) * C_ + c];

  const float s_td  = scal_f[base + 0];
  const float ndynf = scal_f[base + 1];
  const float s_nv  = scal_f[base + 2];
  const float nnvmf = scal_f[base + 3];
  const int   fdyn  = scal_i[base + 4];
  const float denom = fmaxf(1.f, ndynf);
  const float summary =
      (acc + dw[c] * s_td + db[c] * ndynf + vw[c] * s_nv + vb[c] * nnvmf) / denom;

  out[((size_t)b * C_ + c) * T_ + fdyn] = summary;
}

// ---------------------------------------------------------------------------
extern "C" void kernel_launch(void* const* d_in, const int* in_sizes, int n_in,
                              void* d_out, int out_size, void* d_ws, size_t ws_size,
                              hipStream_t stream) {
  const int*           code        = (const int*)d_in[0];
  const float*         nv          = (const float*)d_in[1];
  const float*         td          = (const float*)d_in[2];
  const unsigned char* static_mask = (const unsigned char*)d_in[3];  // jax bool = 1B
  const unsigned char* nvm         = (const unsigned char*)d_in[4];
  const unsigned char* mask        = (const unsigned char*)d_in[5];
  const float*         dw          = (const float*)d_in[6];
  const float*         db          = (const float*)d_in[7];
  const float*         vw          = (const float*)d_in[8];
  const float*         vb          = (const float*)d_in[9];
  const float*         code_table  = (const float*)d_in[10];
  float*               out         = (float*)d_out;

  float* partial = (float*)d_ws;                          // B_*NSPLIT*C_ floats
  float* scal_f  = partial + (size_t)B_ * NSPLIT * C_;    // B_*8 mixed f32/i32
  int*   scal_i  = (int*)scal_f;

  hse_scalars_kernel<<<B_, 256, 0, stream>>>(static_mask, nvm, td, nv, mask,
                                             scal_f, scal_i);
  hse_colsum_kernel<<<B_ * NSPLIT, 512, 0, stream>>>(code, static_mask, code_table,
                                                     partial);
  hse_embed_kernel<<<B_ * 32, 256, 0, stream>>>(code, nv, td, static_mask, nvm,
                                                dw, db, vw, vb, code_table, out);
  hse_inject_kernel<<<B_, 512, 0, stream>>>(partial, dw, db, vw, vb,
                                            scal_f, scal_i, out);
}